// Dezfouli2019GQL_54743653155151
// MI455X (gfx1250) — compile-verified
//
#include <hip/hip_runtime.h>
#include <hip/hip_bf16.h>
#include <math.h>

// ---------------- problem constants (from reference) ----------------
#define T_STEPS 4096
#define B_SIZE  2048
#define N_ACT   2
#define Q_INIT  0.5f

// ---------------- decomposition constants ----------------
#define SEGS    64
#define SEG_LEN (T_STEPS / SEGS)              // 64
#define CHUNK_T 4                             // timesteps staged per LDS chunk
#define BTH     256                           // batch rows per block (8 waves, wave32)
#define ROWF    7                             // floats per (t,b) input record
#define ROWB    (ROWF * 4)                    // 28 bytes
#define ROWSTRIDE ((size_t)B_SIZE * ROWB)     // 57344 bytes between timesteps
#define UNITS_PER_ROW ((BTH * ROWB) / 16)     // 448 b128 units per timestep tile
#define UNITS_PER_CHUNK (CHUNK_T * UNITS_PER_ROW)   // 1792
#define K_PER_THREAD (UNITS_PER_CHUNK / BTH)        // 7 async b128 loads / thread / chunk
#define BUF_FLOATS (CHUNK_T * BTH * ROWF)           // 7168 floats per LDS buffer

static_assert(SEG_LEN == 64, "combine kernel uses 6 squarings for alpha^64");
static_assert(UNITS_PER_CHUNK % BTH == 0, "even async-load distribution");

#define AS1 __attribute__((address_space(1)))
#define AS3 __attribute__((address_space(3)))

typedef int v4i_t __attribute__((ext_vector_type(4)));   // b128 payload type

// ---------------- CDNA5 async global->LDS data mover ----------------
__device__ __forceinline__ void async_copy_b128(const void* gsrc, void* ldst) {
#if __has_builtin(__builtin_amdgcn_global_load_async_to_lds_b128)
  __builtin_amdgcn_global_load_async_to_lds_b128(
      (AS1 v4i_t*)gsrc, (AS3 v4i_t*)ldst, /*offset=*/0, /*cpol=*/0);
#else
  asm volatile("global_load_async_to_lds_b128 %0, %1, off"
               :: "v"((unsigned)(unsigned long long)(AS3 void*)ldst),
                  "v"(gsrc)
               : "memory");
#endif
}

template <int N>
__device__ __forceinline__ void wait_async() {
#if __has_builtin(__builtin_amdgcn_s_wait_asynccnt)
  __builtin_amdgcn_s_wait_asynccnt(N);
#else
  asm volatile("s_wait_asynccnt %0" :: "n"(N) : "memory");
#endif
}

__device__ __forceinline__ float sigmoidf_(float x) {
  return 1.0f / (1.0f + expf(-x));
}

// ---------------- main streaming scan kernel ----------------
// MODE 0: run segment from zero state, write affine constants c_q/c_h to ws
// MODE 1: run segment from ws start state, write logits (+ finals at s==SEGS-1)
// MODE 2: monolithic fallback — whole T from true init, write logits + finals
template <int MODE>
__global__ __launch_bounds__(BTH)
void gql_scan_kernel(const float* __restrict__ inp,
                     const float* __restrict__ phi_logit,
                     const float* __restrict__ chi_logit,
                     const float* __restrict__ beta_p,
                     const float* __restrict__ kappa_p,
                     const float* __restrict__ Cm,
                     float* __restrict__ ws,
                     float* __restrict__ out,
                     int nchunks)
{
  __shared__ float smem[2 * BUF_FLOATS];   // 57344 B, double buffer

  const int tid = threadIdx.x;
  const int s   = blockIdx.y;              // segment (0 for MODE 2)
  const int gb  = blockIdx.x * BTH + tid;  // global batch row
  const int t0  = s * SEG_LEN;

  const char* seg_base = (const char*)inp
                       + (size_t)t0 * ROWSTRIDE
                       + (size_t)blockIdx.x * BTH * ROWB;

  const float phi[2] = { sigmoidf_(phi_logit[0]), sigmoidf_(phi_logit[1]) };
  const float chi[2] = { sigmoidf_(chi_logit[0]), sigmoidf_(chi_logit[1]) };
  const float bet[2] = { beta_p[0], beta_p[1] };
  const float kap[2] = { kappa_p[0], kappa_p[1] };
  const float C00 = Cm[0], C01 = Cm[1], C10 = Cm[2], C11 = Cm[3];

  float* wsq = ws;
  float* wsh = ws + (size_t)SEGS * B_SIZE * 4;
  const int sidx = s * B_SIZE * 4 + gb * 4;

  float q[2][2], h[2][2];
#pragma unroll
  for (int a = 0; a < 2; ++a)
#pragma unroll
    for (int d = 0; d < 2; ++d) {
      if (MODE == 0)      { q[a][d] = 0.0f;                  h[a][d] = 0.0f; }
      else if (MODE == 1) { q[a][d] = wsq[sidx + a * 2 + d]; h[a][d] = wsh[sidx + a * 2 + d]; }
      else                { q[a][d] = Q_INIT;                h[a][d] = 0.0f; }
    }

  auto issue = [&](int c) {
    const char* gchunk = seg_base + (size_t)c * CHUNK_T * ROWSTRIDE;
    char* lbuf = (char*)smem + (size_t)(c & 1) * BUF_FLOATS * 4;
#pragma unroll
    for (int k = 0; k < K_PER_THREAD; ++k) {
      const int u     = tid + k * BTH;            // contiguous 16B units per wave
      const int row   = u / UNITS_PER_ROW;        // timestep within chunk
      const int inrow = u - row * UNITS_PER_ROW;
      async_copy_b128(gchunk + (size_t)row * ROWSTRIDE + (size_t)inrow * 16,
                      lbuf + (size_t)u * 16);
    }
  };

  issue(0);
  for (int c = 0; c < nchunks; ++c) {
    if (c + 1 < nchunks) { issue(c + 1); wait_async<K_PER_THREAD>(); }
    else                 { wait_async<0>(); }
    __syncthreads();     // all waves' chunk-c async writes visible

    const float* buf = smem + (c & 1) * BUF_FLOATS;
#pragma unroll
    for (int tl = 0; tl < CHUNK_T; ++tl) {
      // stride-7 floats across lanes: 7 coprime with 64 banks -> conflict-free
      const float* rec = buf + tl * (BTH * ROWF) + tid * ROWF;
      const float act0 = rec[0], act1 = rec[1];
      const float rew0 = rec[2], rew1 = rec[3];

      float lg[2];
#pragma unroll
      for (int a = 0; a < 2; ++a) {
        const float ch = (a == 0) ? act0 : act1;
        const float rw = (a == 0) ? rew0 : rew1;
#pragma unroll
        for (int d = 0; d < 2; ++d) {
          q[a][d] = (1.0f - phi[d]) * q[a][d] + phi[d] * rw * ch;
          h[a][d] = (1.0f - chi[d]) * h[a][d] + chi[d] * ch;
        }
        lg[a] = bet[0] * q[a][0] + bet[1] * q[a][1]
              + kap[0] * h[a][0] + kap[1] * h[a][1]
              + h[a][0] * (C00 * q[a][0] + C01 * q[a][1])
              + h[a][1] * (C10 * q[a][0] + C11 * q[a][1]);
      }

      if (MODE != 0) {
        const int t = t0 + c * CHUNK_T + tl;
        float2 v;
        if (t == T_STEPS - 1) { v.x = 0.0f; v.y = 0.0f; }   // logits.at[-1].set(0)
        else                  { v.x = lg[0]; v.y = lg[1]; }
        *(float2*)(out + (size_t)t * (B_SIZE * N_ACT) + (size_t)gb * N_ACT) = v;
      }
    }
    __syncthreads();     // protect WAR on buf before chunk c+2 overwrites it
  }

  if (MODE == 0) {
#pragma unroll
    for (int a = 0; a < 2; ++a)
#pragma unroll
      for (int d = 0; d < 2; ++d) {
        wsq[sidx + a * 2 + d] = q[a][d];
        wsh[sidx + a * 2 + d] = h[a][d];
      }
  } else if (MODE == 2 || s == SEGS - 1) {
    float* qf = out + (size_t)T_STEPS * B_SIZE * N_ACT;
    float* hf = qf + (size_t)B_SIZE * N_ACT * 2;
#pragma unroll
    for (int a = 0; a < 2; ++a)
#pragma unroll
      for (int d = 0; d < 2; ++d) {
        qf[gb * 4 + a * 2 + d] = q[a][d];
        hf[gb * 4 + a * 2 + d] = h[a][d];
      }
  }
}

// ---------------- cross-segment affine combine ----------------
// ws holds per-segment constants c (state after segment from zero init).
// Overwrite in place with per-segment START states:
//   start_0 = init;  start_{s+1} = alpha^SEG_LEN * start_s + c_s
__global__ __launch_bounds__(256)
void gql_combine_kernel(const float* __restrict__ phi_logit,
                        const float* __restrict__ chi_logit,
                        float* __restrict__ ws)
{
  const int idx  = blockIdx.x * 256 + threadIdx.x;   // 0 .. 2*B*4 - 1
  const int kind = idx >> 13;                        // 8192 chains per kind
  const int rem  = idx & 8191;                       // b*4 + a*2 + d
  const int d    = rem & 1;

  const float rate = sigmoidf_((kind == 0) ? phi_logit[d] : chi_logit[d]);
  float aL = 1.0f - rate;
#pragma unroll
  for (int i = 0; i < 6; ++i) aL *= aL;              // (1-rate)^64

  float run = (kind == 0) ? Q_INIT : 0.0f;           // Q_INIT / H_INIT
  float* base = ws + (size_t)kind * SEGS * B_SIZE * 4 + rem;
  for (int s = 0; s < SEGS; ++s) {
    const float cval = base[(size_t)s * B_SIZE * 4];
    base[(size_t)s * B_SIZE * 4] = run;              // start state for segment s
    run = aL * run + cval;                           // start state for segment s+1
  }
}

// ---------------- launcher ----------------
extern "C" void kernel_launch(void* const* d_in, const int* in_sizes, int n_in,
                              void* d_out, int out_size, void* d_ws, size_t ws_size,
                              hipStream_t stream) {
  (void)in_sizes; (void)n_in; (void)out_size;
  const float* inp  = (const float*)d_in[0];
  const float* phi  = (const float*)d_in[1];
  const float* chi  = (const float*)d_in[2];
  const float* beta = (const float*)d_in[3];
  const float* kap  = (const float*)d_in[4];
  const float* Cm   = (const float*)d_in[5];
  float* out = (float*)d_out;
  float* ws  = (float*)d_ws;

  const size_t ws_need = (size_t)2 * SEGS * B_SIZE * 4 * sizeof(float);  // 4 MB
  dim3 block(BTH);

  if (ws_size >= ws_need) {
    dim3 grid(B_SIZE / BTH, SEGS);                   // 8 x 64 blocks, 131072 threads
    gql_scan_kernel<0><<<grid, block, 0, stream>>>(inp, phi, chi, beta, kap, Cm,
                                                   ws, out, SEG_LEN / CHUNK_T);
    gql_combine_kernel<<<(2 * B_SIZE * 4) / 256, 256, 0, stream>>>(phi, chi, ws);
    gql_scan_kernel<1><<<grid, block, 0, stream>>>(inp, phi, chi, beta, kap, Cm,
                                                   ws, out, SEG_LEN / CHUNK_T);
  } else {
    dim3 grid(B_SIZE / BTH, 1);                      // fallback: single full-T pass
    gql_scan_kernel<2><<<grid, block, 0, stream>>>(inp, phi, chi, beta, kap, Cm,
                                                   ws, out, T_STEPS / CHUNK_T);
  }
}